// IMUIntegrator_30047591202942
// MI455X (gfx1250) — compile-verified
//
#include <hip/hip_runtime.h>
#include <cstdint>

#define GRAV 9.81007f

#if defined(__AMDGCN__) && __has_builtin(__builtin_amdgcn_wmma_f32_16x16x4_f32)
#define HAS_WMMA 1
#else
#define HAS_WMMA 0
#endif

#if defined(__AMDGCN__) && __has_builtin(__builtin_amdgcn_global_load_async_to_lds_b32) && \
    __has_builtin(__builtin_amdgcn_global_load_async_to_lds_b128)
#define HAS_ASYNC 1
#else
#define HAS_ASYNC 0
#endif

typedef float v2f __attribute__((ext_vector_type(2)));
typedef float v8f __attribute__((ext_vector_type(8)));

struct Q  { float x, y, z, w; };
struct V3 { float x, y, z; };
struct St { Q q; V3 v; V3 p; float t; };

__device__ __forceinline__ V3 v3add(V3 a, V3 b){ return {a.x+b.x, a.y+b.y, a.z+b.z}; }
__device__ __forceinline__ V3 v3sub(V3 a, V3 b){ return {a.x-b.x, a.y-b.y, a.z-b.z}; }
__device__ __forceinline__ V3 v3scale(V3 a, float s){ return {a.x*s, a.y*s, a.z*s}; }
__device__ __forceinline__ V3 v3cross(V3 a, V3 b){
  return {a.y*b.z - a.z*b.y, a.z*b.x - a.x*b.z, a.x*b.y - a.y*b.x};
}
// Hamilton product, matches reference qmul: v = w1*v2 + w2*v1 + v1 x v2
__device__ __forceinline__ Q qmul(Q a, Q b){
  Q r;
  r.w = a.w*b.w - (a.x*b.x + a.y*b.y + a.z*b.z);
  r.x = a.w*b.x + b.w*a.x + (a.y*b.z - a.z*b.y);
  r.y = a.w*b.y + b.w*a.y + (a.z*b.x - a.x*b.z);
  r.z = a.w*b.z + b.w*a.z + (a.x*b.y - a.y*b.x);
  return r;
}
__device__ __forceinline__ V3 qrotv(Q q, V3 v){
  V3 qv = {q.x, q.y, q.z};
  V3 t = v3add(v3cross(qv, v), v3scale(v, q.w));
  return v3add(v, v3scale(v3cross(qv, t), 2.0f));
}
// Associative combine of integration states (A earlier in time than B)
__device__ __forceinline__ St comb(const St& A, const St& B){
  St r;
  r.q = qmul(A.q, B.q);
  r.v = v3add(A.v, qrotv(A.q, B.v));
  r.p = v3add(A.p, v3add(v3scale(A.v, B.t), qrotv(A.q, B.p)));
  r.t = A.t + B.t;
  return r;
}
__device__ __forceinline__ St st_ident(){ St s = {{0,0,0,1},{0,0,0},{0,0,0},0.f}; return s; }

__device__ __forceinline__ float shup(float v, int d){ return __shfl_up(v, d, 32); }
__device__ __forceinline__ float shbc(float v, int l){ return __shfl(v, l, 32); }
__device__ __forceinline__ Q shup_q(Q q, int d){
  Q r; r.x = shup(q.x,d); r.y = shup(q.y,d); r.z = shup(q.z,d); r.w = shup(q.w,d); return r;
}
__device__ __forceinline__ St shup_st(const St& s, int d){
  St r; r.q = shup_q(s.q,d);
  r.v = { shup(s.v.x,d), shup(s.v.y,d), shup(s.v.z,d) };
  r.p = { shup(s.p.x,d), shup(s.p.y,d), shup(s.p.z,d) };
  r.t = shup(s.t,d); return r;
}
__device__ __forceinline__ St shbc_st(const St& s, int l){
  St r;
  r.q = { shbc(s.q.x,l), shbc(s.q.y,l), shbc(s.q.z,l), shbc(s.q.w,l) };
  r.v = { shbc(s.v.x,l), shbc(s.v.y,l), shbc(s.v.z,l) };
  r.p = { shbc(s.p.x,l), shbc(s.p.y,l), shbc(s.p.z,l) };
  r.t = shbc(s.t,l); return r;
}

__device__ __forceinline__ void lds_fence(){
#if defined(__AMDGCN__)
  __builtin_amdgcn_wave_barrier();
#if __has_builtin(__builtin_amdgcn_s_wait_dscnt)
  __builtin_amdgcn_s_wait_dscnt(0);
#else
  asm volatile("s_wait_dscnt 0" ::: "memory");
#endif
  __builtin_amdgcn_wave_barrier();
#endif
}

#if HAS_ASYNC
// Builtin signatures (from hipcc diagnostics):
//   b32 : (int AS1*, int AS1/AS3*, imm, imm)
//   b128: (int __vector(4) AS1*, ... AS3*, imm, imm)
typedef int vint4 __attribute__((vector_size(16)));
typedef __attribute__((address_space(1))) int   GInt;
typedef __attribute__((address_space(3))) int   LInt;
typedef __attribute__((address_space(1))) vint4 GInt4;
typedef __attribute__((address_space(3))) vint4 LInt4;

__device__ __forceinline__ void asy_b32(const float* g, const float* l){
  __builtin_amdgcn_global_load_async_to_lds_b32(
      (GInt*)(uintptr_t)g, (LInt*)(uint32_t)(uintptr_t)l, 0, 0);
}
__device__ __forceinline__ void asy_b128(const float* g, const float* l){
  __builtin_amdgcn_global_load_async_to_lds_b128(
      (GInt4*)(uintptr_t)g, (LInt4*)(uint32_t)(uintptr_t)l, 0, 0);
}
#if __has_builtin(__builtin_amdgcn_s_wait_asynccnt)
#define WAIT_ASYNC(n) do { __builtin_amdgcn_s_wait_asynccnt(n); asm volatile("" ::: "memory"); } while(0)
#else
#define WAIT_ASYNC(n) asm volatile("s_wait_asynccnt %0" :: "i"(n) : "memory")
#endif
// Stage one 32-step chunk (dt:32f | gyro:96f | acc:96f | gt_rot:128f = 352 floats)
__device__ __forceinline__ void issue_chunk(const float* dtg, const float* gyg,
                                            const float* acg, const float* qg,
                                            long long base, int lane, float* buf){
  asy_b32 (dtg + base        + lane,      buf +   0 + lane);
  asy_b32 (gyg + base*3      + lane,      buf +  32 + lane);
  asy_b32 (gyg + base*3 + 32 + lane,      buf +  64 + lane);
  asy_b32 (gyg + base*3 + 64 + lane,      buf +  96 + lane);
  asy_b32 (acg + base*3      + lane,      buf + 128 + lane);
  asy_b32 (acg + base*3 + 32 + lane,      buf + 160 + lane);
  asy_b32 (acg + base*3 + 64 + lane,      buf + 192 + lane);
  asy_b128(qg  + base*4      + lane*4,    buf + 224 + lane*4);
}
#else
#define WAIT_ASYNC(n)
#endif

// Inclusive cumsum over the 32 lanes for NCH (<=8) channels.
// WMMA path: Y = L16 * X via 4 chained v_wmma_f32_16x16x4_f32; both 16-lane
// groups share the 16 N-columns (cols 0-7 = group0 ch, cols 8-15 = group1 ch),
// then group1 adds group0's totals. A (16x4 f32): lane l -> M=l%16,
// K=r+2*(l>=16). B (4x16): lane l -> N=l%16, K=r+2*(l>=16). D: lane l ->
// M=r+8*(l>=16), N=l%16 (per CDNA5 ISA layouts).
template<int NCH>
__device__ __forceinline__ void wave_cumsum(float* x, int lane,
                                            float (*xb)[9], float (*yb)[17],
                                            const v2f* As){
#if HAS_WMMA
  #pragma unroll
  for (int c = 0; c < NCH; ++c) xb[lane][c] = x[c];
  #pragma unroll
  for (int c = NCH; c < 8; ++c) xb[lane][c] = 0.f;
  lds_fence();
  v8f acc = {0.f,0.f,0.f,0.f,0.f,0.f,0.f,0.f};
  const int N     = lane & 15;
  const int c0    = lane & 7;
  const int gsel  = N >> 3;            // which 16-step group this column reads
  const int khalf = (lane >> 4) * 2;   // K offset for upper half-wave
  #pragma unroll
  for (int kk = 0; kk < 4; ++kk) {
    v2f b;
    b.x = xb[gsel*16 + kk*4 + khalf + 0][c0];
    b.y = xb[gsel*16 + kk*4 + khalf + 1][c0];
    acc = __builtin_amdgcn_wmma_f32_16x16x4_f32(false, As[kk], false, b,
                                                (short)0, acc, false, false);
  }
  const int mo = (lane >> 4) * 8;
  #pragma unroll
  for (int r = 0; r < 8; ++r) yb[mo + r][N] = acc[r];
  lds_fence();
  const int s = lane & 15, grp = lane >> 4;
  #pragma unroll
  for (int c = 0; c < NCH; ++c) {
    float v = yb[s][grp*8 + c];
    if (grp) v += yb[15][c];           // add group0 totals into group1
    x[c] = v;
  }
#else
  (void)xb; (void)yb; (void)As;
  #pragma unroll
  for (int sft = 1; sft < 32; sft <<= 1) {
    float t[NCH];
    #pragma unroll
    for (int c = 0; c < NCH; ++c) t[c] = shup(x[c], sft);
    if (lane >= sft) {
      #pragma unroll
      for (int c = 0; c < NCH; ++c) x[c] += t[c];
    }
  }
#endif
}

#define NW 4  // waves per block

// Pass1 (WRITE=false): per-segment totals.  Pass3 (WRITE=true): apply carries,
// write rot/vel/pos.
template<bool WRITE>
__global__ __launch_bounds__(NW*32)
void imu_pass(const float* __restrict__ dtg, const float* __restrict__ gyg,
              const float* __restrict__ acg, const float* __restrict__ qg,
              const float* __restrict__ carries, float* __restrict__ totals,
              const float* __restrict__ irot, const float* __restrict__ ivel,
              const float* __restrict__ ipos, float* __restrict__ out,
              long long NF, int F, int S, int segLen, int nChunks, int nWaves){
#if HAS_ASYNC
  __shared__ __align__(16) float stage[NW][2][352];
#endif
#if HAS_WMMA
  __shared__ float xbS[NW][32][9];
  __shared__ float ybS[NW][16][17];
#endif
  const int wv = threadIdx.x >> 5, lane = threadIdx.x & 31;
  const long long wid = (long long)blockIdx.x * NW + wv;
  if (wid >= nWaves) return;
  const int b = (int)(wid / S), seg = (int)(wid % S);
  const long long base0 = (long long)b * F + (long long)seg * segLen;

#if HAS_WMMA
  float (*xbp)[9]  = xbS[wv];
  float (*ybp)[17] = ybS[wv];
#else
  float (*xbp)[9]  = nullptr;
  float (*ybp)[17] = nullptr;
#endif
  // Constant lower-triangular A slices (16x4 each), per ISA A-layout.
  v2f As[4];
  {
    const int M = lane & 15, kh = (lane >> 4) * 2;
    #pragma unroll
    for (int kk = 0; kk < 4; ++kk) {
      As[kk].x = (kk*4 + kh + 0 <= M) ? 1.f : 0.f;
      As[kk].y = (kk*4 + kh + 1 <= M) ? 1.f : 0.f;
    }
  }

  St carry;
  Q qi; V3 vi, pi;
  if constexpr (WRITE) {
    const float* cw = carries + (size_t)wid * 12;
    carry.q = { cw[0], cw[1], cw[2], cw[3] };
    carry.v = { cw[4], cw[5], cw[6] };
    carry.p = { cw[7], cw[8], cw[9] };
    carry.t = cw[10];
    qi = { irot[b*4+0], irot[b*4+1], irot[b*4+2], irot[b*4+3] };
    vi = { ivel[b*3+0], ivel[b*3+1], ivel[b*3+2] };
    pi = { ipos[b*3+0], ipos[b*3+1], ipos[b*3+2] };
  } else {
    carry = st_ident();
  }

#if HAS_ASYNC
  issue_chunk(dtg, gyg, acg, qg, base0, lane, stage[wv][0]);
#endif

  for (int ch = 0; ch < nChunks; ++ch) {
    const long long base = base0 + (long long)ch * 32;
    float dt; V3 gy, ac; Q gt;
#if HAS_ASYNC
    if (ch + 1 < nChunks) {
      issue_chunk(dtg, gyg, acg, qg, base + 32, lane, stage[wv][(ch+1)&1]);
      WAIT_ASYNC(8);   // the 8 loads of the current chunk have completed
    } else {
      WAIT_ASYNC(0);
    }
    __builtin_amdgcn_wave_barrier();
    const float* sb = stage[wv][ch & 1];
    dt = sb[lane];
    gy = { sb[ 32+lane*3], sb[ 33+lane*3], sb[ 34+lane*3] };
    ac = { sb[128+lane*3], sb[129+lane*3], sb[130+lane*3] };
    gt = { sb[224+lane*4], sb[225+lane*4], sb[226+lane*4], sb[227+lane*4] };
#else
    const long long e = base + lane;
    dt = dtg[e];
    gy = { gyg[e*3], gyg[e*3+1], gyg[e*3+2] };
    ac = { acg[e*3], acg[e*3+1], acg[e*3+2] };
    gt = { qg[e*4], qg[e*4+1], qg[e*4+2], qg[e*4+3] };
    if (ch + 1 < nChunks) {
      __builtin_prefetch(dtg + e + 32, 0, 1);
      __builtin_prefetch(gyg + (e+32)*3, 0, 1);
      __builtin_prefetch(acg + (e+32)*3, 0, 1);
      __builtin_prefetch(qg  + (e+32)*4, 0, 1);
    }
#endif
    // dr = so3_exp(gyro*dt)
    V3 phi = v3scale(gy, dt);
    float th2 = phi.x*phi.x + phi.y*phi.y + phi.z*phi.z;
    float th  = sqrtf(th2);
    float k   = (th < 1e-6f) ? (0.5f - th2 * (1.f/48.f)) : (sinf(0.5f*th)/th);
    Q dr = { phi.x*k, phi.y*k, phi.z*k, cosf(0.5f*th) };
    // a = acc - qrot(qinv(gt_rot), g)
    Q gtc = { -gt.x, -gt.y, -gt.z, gt.w };
    V3 gvec = {0.f, 0.f, GRAV};
    V3 a = v3sub(ac, qrotv(gtc, gvec));
    // 32-wide inclusive quaternion scan (non-commutative, log steps)
    Q lq = dr;
    #pragma unroll
    for (int sft = 1; sft < 32; sft <<= 1) {
      Q t = shup_q(lq, sft);
      if (lane >= sft) lq = qmul(t, lq);
    }
    V3 ra = qrotv(lq, a);
    // inclusive cumsum of (ra*dt, dt) -> localV, localT
    float x4[4] = { ra.x*dt, ra.y*dt, ra.z*dt, dt };
    wave_cumsum<4>(x4, lane, xbp, ybp, As);
    V3 lV = { x4[0], x4[1], x4[2] };
    float lT = x4[3];
    V3 lVex = v3sub(lV, v3scale(ra, dt));      // exclusive V
    float hdt2 = 0.5f * dt * dt;
    float x3[3] = { lVex.x*dt + ra.x*hdt2, lVex.y*dt + ra.y*hdt2, lVex.z*dt + ra.z*hdt2 };
    wave_cumsum<3>(x3, lane, xbp, ybp, As);
    St local = { lq, lV, { x3[0], x3[1], x3[2] }, lT };
    St g = comb(carry, local);                 // global (pass3) / in-segment (pass1)
    if constexpr (WRITE) {
      const long long idx = base + lane;
      Q rot = qmul(qi, g.q);
      V3 vel = v3add(vi, qrotv(qi, g.v));
      V3 pos = v3add(pi, v3add(v3scale(vi, g.t), qrotv(qi, g.p)));
      float4 r4; r4.x = rot.x; r4.y = rot.y; r4.z = rot.z; r4.w = rot.w;
      reinterpret_cast<float4*>(out)[idx] = r4;
      float* vp = out + (size_t)NF*4 + (size_t)idx*3;
      vp[0] = vel.x; vp[1] = vel.y; vp[2] = vel.z;
      float* pp = out + (size_t)NF*7 + (size_t)idx*3;
      pp[0] = pos.x; pp[1] = pos.y; pp[2] = pos.z;
    }
    carry = shbc_st(g, 31);                    // chunk carry for next iteration
  }
  if constexpr (!WRITE) {
    if (lane == 0) {
      float* o = totals + (size_t)wid * 12;
      o[0]=carry.q.x; o[1]=carry.q.y; o[2]=carry.q.z; o[3]=carry.q.w;
      o[4]=carry.v.x; o[5]=carry.v.y; o[6]=carry.v.z;
      o[7]=carry.p.x; o[8]=carry.p.y; o[9]=carry.p.z;
      o[10]=carry.t;  o[11]=0.f;
    }
  }
}

// Pass 2: one wave per batch scans S segment totals -> exclusive carries.
__global__ __launch_bounds__(32)
void imu_seg_scan(const float* __restrict__ totals, float* __restrict__ carries, int S){
  const int b = blockIdx.x, lane = threadIdx.x;
  St carry = st_ident();
  for (int base = 0; base < S; base += 32) {
    const int i = base + lane;
    const bool ok = i < S;
    St s = st_ident();
    if (ok) {
      const float* t = totals + ((size_t)b * S + i) * 12;
      s.q = { t[0], t[1], t[2], t[3] };
      s.v = { t[4], t[5], t[6] };
      s.p = { t[7], t[8], t[9] };
      s.t = t[10];
    }
    #pragma unroll
    for (int sft = 1; sft < 32; sft <<= 1) {
      St t = shup_st(s, sft);
      if (lane >= sft) s = comb(t, s);
    }
    St ex = shup_st(s, 1);
    St mine = (lane == 0) ? carry : comb(carry, ex);
    if (ok) {
      float* o = carries + ((size_t)b * S + i) * 12;
      o[0]=mine.q.x; o[1]=mine.q.y; o[2]=mine.q.z; o[3]=mine.q.w;
      o[4]=mine.v.x; o[5]=mine.v.y; o[6]=mine.v.z;
      o[7]=mine.p.x; o[8]=mine.p.y; o[9]=mine.p.z;
      o[10]=mine.t;  o[11]=0.f;
    }
    St last = shbc_st(s, 31);
    carry = comb(carry, last);
  }
}

extern "C" void kernel_launch(void* const* d_in, const int* in_sizes, int n_in,
                              void* d_out, int out_size, void* d_ws, size_t ws_size,
                              hipStream_t stream) {
  const float* dt = (const float*)d_in[0];
  const float* gy = (const float*)d_in[1];
  const float* ac = (const float*)d_in[2];
  const float* gq = (const float*)d_in[3];
  const float* ir = (const float*)d_in[4];
  const float* iv = (const float*)d_in[5];
  const float* ip = (const float*)d_in[6];
  float* out = (float*)d_out;

  int B = in_sizes[4] / 4; if (B <= 0) B = 1;
  int F = in_sizes[0] / B;
  int S = 128;
  while (S > 1 && (F % (S * 32) != 0)) S >>= 1;
  const int segLen  = F / S;
  const int nChunks = segLen / 32;
  const long long NF = (long long)B * F;

  float* totals  = (float*)d_ws;
  float* carries = totals + (size_t)B * S * 12;

  const int nWaves = B * S;
  const int blocks = (nWaves + NW - 1) / NW;

  imu_pass<false><<<blocks, NW * 32, 0, stream>>>(
      dt, gy, ac, gq, nullptr, totals, nullptr, nullptr, nullptr, nullptr,
      NF, F, S, segLen, nChunks, nWaves);
  imu_seg_scan<<<B, 32, 0, stream>>>(totals, carries, S);
  imu_pass<true><<<blocks, NW * 32, 0, stream>>>(
      dt, gy, ac, gq, carries, nullptr, ir, iv, ip, out,
      NF, F, S, segLen, nChunks, nWaves);
}